// LightGCN_86182813762458
// MI455X (gfx1250) — compile-verified
//
#include <hip/hip_runtime.h>

typedef float v2f __attribute__((ext_vector_type(2)));
typedef float v8f __attribute__((ext_vector_type(8)));

#define LGCN_EPS 1e-12f

// ---------------- zero fill ----------------
__global__ void lgcn_zero_kernel(float* __restrict__ p, long n) {
  long i = (long)blockIdx.x * blockDim.x + threadIdx.x;
  long stride = (long)gridDim.x * blockDim.x;
  for (; i < n; i += stride) p[i] = 0.0f;
}

// ---------------- SpMM scatter: y[rows[e]] += vals[e] * x[cols[e]] --------
// wave32 per edge; each lane owns 2 of the 64 dims (b64 gather, 2x f32 atomics)
__global__ __launch_bounds__(256) void lgcn_spmm_kernel(
    const float* __restrict__ x,
    const int* __restrict__ rows,
    const int* __restrict__ cols,
    const float* __restrict__ vals,
    float* __restrict__ y, int E) {
  int lane = threadIdx.x & 31;
  int warp = (int)((blockIdx.x * blockDim.x + threadIdx.x) >> 5);
  int nwarp = (int)((gridDim.x * blockDim.x) >> 5);
  for (int e = warp; e < E; e += nwarp) {
    int r = rows[e];
    int c = cols[e];
    float v = vals[e];
    float2 xv = *((const float2*)(x + (size_t)c * 64) + lane);
    float* yp = y + (size_t)r * 64 + lane * 2;
    atomicAdd(yp, v * xv.x);
    atomicAdd(yp + 1, v * xv.y);
  }
}

// select element i (0..7) of a v8f (cndmask chain)
__device__ __forceinline__ float lgcn_pick8(v8f c, int i) {
  float r = c[0];
  if (i == 1) r = c[1];
  if (i == 2) r = c[2];
  if (i == 3) r = c[3];
  if (i == 4) r = c[4];
  if (i == 5) r = c[5];
  if (i == 6) r = c[6];
  if (i == 7) r = c[7];
  return r;
}

// ---------------- per-16-row post pass ----------------
// Gram diagonals via V_WMMA_F32_16X16X4_F32 (exact f32):
//   dxx = ||x_row||^2, dyy = ||y_row||^2, dxy = x_row . y_row
// then x2 = exp(1/(3*relu(cos) + adj*(1+layer))); slot *= x2 (in place),
// x <- y (next-layer input), optionally final tail <- y.
__global__ __launch_bounds__(256) void lgcn_post_kernel(
    float* __restrict__ y,        // raw SpMM result (d_out slot), scaled in place
    float* __restrict__ x,        // previous x (ws); overwritten with y
    const float* __restrict__ adj_n,
    float* __restrict__ x_final,  // may be null
    int N, int layer) {
  int lane = threadIdx.x & 31;
  int tile = (int)((blockIdx.x * blockDim.x + threadIdx.x) >> 5);
  int ntiles = N >> 4;
  if (tile >= ntiles) return;  // wave-uniform exit: surviving waves have EXEC all-1

  int R = tile << 4;
  int mrow = R + (lane & 15);
  int kh = (lane >> 4) << 1;  // lanes 16..31 hold K+2,K+3 (A layout, 16x4 f32)
  const float* xr = x + (size_t)mrow * 64 + kh;
  const float* yr = y + (size_t)mrow * 64 + kh;

  v8f cxx = {}; v8f cyy = {}; v8f cxy = {};
#pragma unroll
  for (int kk = 0; kk < 64; kk += 4) {
    float2 fx = *(const float2*)(xr + kk);
    float2 fy = *(const float2*)(yr + kk);
    v2f ax; ax[0] = fx.x; ax[1] = fx.y;
    v2f ay; ay[0] = fy.x; ay[1] = fy.y;
    // A-layout of a 16x4 f32 tile == B-layout of its transpose, so
    // wmma(ax, ay) accumulates the 16x16 cross-Gram sum_k X[m,k]*Y[n,k].
    cxx = __builtin_amdgcn_wmma_f32_16x16x4_f32(false, ax, false, ax,
                                                (short)0, cxx, false, false);
    cyy = __builtin_amdgcn_wmma_f32_16x16x4_f32(false, ay, false, ay,
                                                (short)0, cyy, false, false);
    cxy = __builtin_amdgcn_wmma_f32_16x16x4_f32(false, ax, false, ay,
                                                (short)0, cxy, false, false);
  }

  // Diagonal lives at: lanes 0..7 -> row = lane, elem = lane
  //                    lanes 24..31 -> row = lane-16, elem = lane-24
  int idx = ((lane < 8) ? lane : (lane - 24)) & 7;
  float dxx = lgcn_pick8(cxx, idx);
  float dyy = lgcn_pick8(cyy, idx);
  float dxy = lgcn_pick8(cxy, idx);
  int myrow = ((lane < 8) ? lane : (lane - 16)) & 15;  // valid on lanes 0-7,24-31
  float adj = adj_n[R + myrow];

  float nx = fmaxf(sqrtf(fmaxf(dxx, 0.0f)), LGCN_EPS);
  float ny = fmaxf(sqrtf(fmaxf(dyy, 0.0f)), LGCN_EPS);
  float cosv = fmaxf(dxy / (nx * ny), 0.0f);
  float x2 = expf(1.0f / (3.0f * cosv + adj * (1.0f + (float)layer)));

#pragma unroll
  for (int rr = 0; rr < 16; ++rr) {
    int src = (rr < 8) ? rr : (rr + 16);  // lane holding row rr's x2
    float s = __shfl(x2, src, 32);
    size_t rowoff = (size_t)(R + rr) * 64;
    float2 yv = *((const float2*)(y + rowoff) + lane);
    *((float2*)(x + rowoff) + lane) = yv;           // next-layer input (raw)
    if (x_final) *((float2*)(x_final + rowoff) + lane) = yv;
    float2 o;
    o.x = yv.x * s;
    o.y = yv.y * s;
    *((float2*)(y + rowoff) + lane) = o;            // all_emb[i] = y * x2
  }
}

extern "C" void kernel_launch(void* const* d_in, const int* in_sizes, int n_in,
                              void* d_out, int out_size, void* d_ws, size_t ws_size,
                              hipStream_t stream) {
  const float* user_emb = (const float*)d_in[0];
  const float* item_emb = (const float*)d_in[1];
  const int* rows = (const int*)d_in[2];
  const int* cols = (const int*)d_in[3];
  const float* vals = (const float*)d_in[4];
  const float* adj_n = (const float*)d_in[5];

  int nU = in_sizes[0] / 64;
  int nI = in_sizes[1] / 64;
  int N = nU + nI;
  int E = in_sizes[2];
  long nd = (long)N * 64;
  int layers = (int)((long)out_size / nd) - 1;  // output = [layers, N, 64] + [N, 64]

  float* x = (float*)d_ws;       // N*64 floats of scratch
  float* out = (float*)d_out;

  // x = concat(user_emb, item_emb)
  hipMemcpyAsync(x, user_emb, (size_t)in_sizes[0] * sizeof(float),
                 hipMemcpyDeviceToDevice, stream);
  hipMemcpyAsync(x + in_sizes[0], item_emb, (size_t)in_sizes[1] * sizeof(float),
                 hipMemcpyDeviceToDevice, stream);

  int ntiles = N / 16;
  for (int i = 0; i < layers; ++i) {
    float* slot = out + (size_t)i * nd;
    lgcn_zero_kernel<<<2048, 256, 0, stream>>>(slot, nd);
    int spmm_blocks = (E + 7) / 8;  // 8 waves/block, one edge per wave
    lgcn_spmm_kernel<<<spmm_blocks, 256, 0, stream>>>(x, rows, cols, vals, slot, E);
    float* xfin = (i == layers - 1) ? (out + (size_t)layers * nd) : nullptr;
    lgcn_post_kernel<<<(ntiles + 7) / 8, 256, 0, stream>>>(slot, x, adj_n, xfin, N, i);
  }
}